// HashEmbedder_79413945303930
// MI455X (gfx1250) — compile-verified
//
#include <hip/hip_runtime.h>

// Multiresolution hash-grid embedding lookup (Instant-NGP style) for MI455X.
//
// Gather/binary-search bound; all tables+embeddings (~80MB) are L2-resident
// (192MB L2), so the optimization target is probe COUNT and latency hiding,
// not HBM bandwidth (out = 128MB -> ~6us at 23.3TB/s, negligible).
//
// Per block: level L = blockIdx.y. A 4096-entry strided sample of the sorted
// table lives in LDS (16KB). Each thread owns one point:
//   phase 1: 13-step branchless lower_bound over LDS samples (4 searches
//            interleaved for ILP) -> window of width S = ceil(n/4096)
//   phase 2: ~log2(S) predicated global probes inside the window
// Only the 4 corners with i=0 are searched; the +x corners are derived
// exactly via lower_bound(h+1) = lower_bound(h) + (table[lb(h)] == h),
// valid because the non-padding table region is strictly increasing
// (np.unique) and h+1 < INT32_MAX (padding value) always.

#define NSAMP      4096
#define NSAMP_LOG2 12
#define NLEVELS    16
#define FDIM       2
#define BLOCK      256

__global__ __launch_bounds__(BLOCK)
void hashgrid_lookup_kernel(const float* __restrict__ x,      // [npoints,3]
                            const float* __restrict__ emb,    // [16,n,2]
                            const int*   __restrict__ ht,     // [16,n] sorted asc (pad INT_MAX)
                            const int*   __restrict__ res,    // [16]
                            float* __restrict__ out,          // [npoints,32]
                            int n, int npoints)
{
    __shared__ int smp[NSAMP];

    const int L = blockIdx.y;
    const int*   __restrict__ table = ht  + (size_t)L * (size_t)n;
    const float* __restrict__ embL  = emb + (size_t)L * (size_t)n * FDIM;

    // Sample stride; same for every level (shared padded length n).
    const int S = (n + NSAMP - 1) >> NSAMP_LOG2;   // ceil(n/4096), >= 1

    // Cooperative LDS fill: smp[i] = table[min((i+1)*S - 1, n-1)]
    for (int i = threadIdx.x; i < NSAMP; i += BLOCK) {
        int idx = (i + 1) * S - 1;
        if (idx > n - 1) idx = n - 1;
        smp[i] = table[idx];
    }
    __syncthreads();

    const int p = blockIdx.x * BLOCK + threadIdx.x;
    if (p >= npoints) return;

    const int   r  = res[L];              // uniform per block (SGPR)
    const float rf = (float)r;
    const int   r2 = r * r;

    // Load + clip point, compute cell + weights exactly like the reference.
    float x0 = x[3 * p + 0];
    float x1 = x[3 * p + 1];
    float x2 = x[3 * p + 2];
    x0 = fminf(fmaxf(x0, -1.0f), 1.0f);
    x1 = fminf(fmaxf(x1, -1.0f), 1.0f);
    x2 = fminf(fmaxf(x2, -1.0f), 1.0f);

    const float grid = 2.0f / rf;         // (BOX_MAX-BOX_MIN)/rf
    const float t0 = (x0 + 1.0f) / grid;
    const float t1 = (x1 + 1.0f) / grid;
    const float t2 = (x2 + 1.0f) / grid;
    const float b0 = floorf(t0), b1 = floorf(t1), b2 = floorf(t2);
    const float wx = t0 - b0, wy = t1 - b1, wz = t2 - b2;
    const int   i0 = (int)b0, i1 = (int)b1, i2 = (int)b2;

    // Hashes for the 4 i=0 corners; c encodes (j,k): c = j*2 + k.
    // Reference corner index is i*4 + j*2 + k, so these are ve[0..3] and the
    // derived +1 hashes are ve[4..7].
    const int hbase = i0 + i1 * r + i2 * r2;
    int h[4];
    h[0] = hbase;               // (j=0,k=0)
    h[1] = hbase + r2;          // (j=0,k=1)
    h[2] = hbase + r;           // (j=1,k=0)
    h[3] = hbase + r + r2;      // (j=1,k=1)

    // ---- Phase 1: branchless lower_bound over NSAMP LDS samples ----
    int pos1[4];
#pragma unroll
    for (int c = 0; c < 4; ++c) pos1[c] = 0;
#pragma unroll
    for (int step = NSAMP; step > 0; step >>= 1) {
#pragma unroll
        for (int c = 0; c < 4; ++c) {
            int nxt = pos1[c] + step;
            int idx = nxt - 1;
            if (idx > NSAMP - 1) idx = NSAMP - 1;      // clamped safe LDS read
            int v = smp[idx];
            if ((nxt <= NSAMP) && (v < h[c])) pos1[c] = nxt;
        }
    }

    // ---- Phase 2: finish lower_bound in global within [lo, lo+S) ----
    int lo[4], n2[4], q[4];
#pragma unroll
    for (int c = 0; c < 4; ++c) {
        lo[c] = pos1[c] * S;
        int rem = n - lo[c];
        if (rem > S) rem = S;
        if (rem < 0) rem = 0;
        n2[c] = rem;
        q[c]  = 0;
    }
    // Largest power-of-two <= S; uniform scalar -> non-divergent loop.
    const int P2 = 1 << (31 - __clz(S));
    for (int step = P2; step > 0; step >>= 1) {
#pragma unroll
        for (int c = 0; c < 4; ++c) {
            int nxt = q[c] + step;
            int gi  = lo[c] + nxt - 1;
            if (gi > n - 1) gi = n - 1;                // clamped safe global read
            int v = table[gi];
            if ((nxt <= n2[c]) && (v < h[c])) q[c] = nxt;
        }
    }

    // ---- Derive the +x corners: lb(h+1) = lb(h) + (table[lb(h)] == h) ----
    int pcl[4], pch[4];   // clipped gather indices for corners c and c+4
#pragma unroll
    for (int c = 0; c < 4; ++c) {
        int pf = lo[c] + q[c];                 // searchsorted(table, h[c]), may be n
        int gi = pf > n - 1 ? n - 1 : pf;
        int v  = table[gi];                    // line just touched by the search
        int pp = pf + ((v == h[c]) ? 1 : 0);   // searchsorted(table, h[c]+1)
        pcl[c] = gi;                           // clip(pf, 0, n-1)
        pch[c] = pp > n - 1 ? n - 1 : pp;      // clip(pp, 0, n-1)
    }

    // ---- Gather embeddings (float2 per corner) ----
    float e0[8], e1[8];
#pragma unroll
    for (int c = 0; c < 4; ++c) {
        const float2 vl = *(const float2*)(embL + (size_t)pcl[c] * FDIM);
        const float2 vh = *(const float2*)(embL + (size_t)pch[c] * FDIM);
        e0[c]     = vl.x;  e1[c]     = vl.y;   // corner c     (i=0)
        e0[c + 4] = vh.x;  e1[c + 4] = vh.y;   // corner c + 4 (i=1)
    }

    // ---- Trilinear interpolation (reference ordering) ----
    const float omx = 1.0f - wx, omy = 1.0f - wy, omz = 1.0f - wz;
    // feature 0
    float c00 = e0[0] * omx + e0[4] * wx;
    float c01 = e0[1] * omx + e0[5] * wx;
    float c10 = e0[2] * omx + e0[6] * wx;
    float c11 = e0[3] * omx + e0[7] * wx;
    float ca  = c00 * omy + c10 * wy;
    float cb  = c01 * omy + c11 * wy;
    const float o0 = ca * omz + cb * wz;
    // feature 1
    c00 = e1[0] * omx + e1[4] * wx;
    c01 = e1[1] * omx + e1[5] * wx;
    c10 = e1[2] * omx + e1[6] * wx;
    c11 = e1[3] * omx + e1[7] * wx;
    ca  = c00 * omy + c10 * wy;
    cb  = c01 * omy + c11 * wy;
    const float o1 = ca * omz + cb * wz;

    // out[p][L*2 + f], coalesced b64 store
    float2 ov; ov.x = o0; ov.y = o1;
    *(float2*)(out + (size_t)p * (NLEVELS * FDIM) + (size_t)L * FDIM) = ov;
}

extern "C" void kernel_launch(void* const* d_in, const int* in_sizes, int n_in,
                              void* d_out, int out_size, void* d_ws, size_t ws_size,
                              hipStream_t stream) {
    (void)n_in; (void)out_size; (void)d_ws; (void)ws_size;
    const float* x   = (const float*)d_in[0];   // [npoints,3] f32
    const float* emb = (const float*)d_in[1];   // [16,n,2]    f32
    const int*   ht  = (const int*)d_in[2];     // [16,n]      i32
    const int*   res = (const int*)d_in[3];     // [16]        i32
    float* out = (float*)d_out;                 // [npoints,32] f32

    const int npoints = in_sizes[0] / 3;
    const int n       = in_sizes[2] / NLEVELS;

    dim3 grid((npoints + BLOCK - 1) / BLOCK, NLEVELS);
    hashgrid_lookup_kernel<<<grid, BLOCK, 0, stream>>>(x, emb, ht, res, out, n, npoints);
}